// QuanvolutionHybridModel_65481071405222
// MI455X (gfx1250) — compile-verified
//
#include <hip/hip_runtime.h>
#include <hip/hip_bf16.h>
#include <math.h>

typedef float v2f __attribute__((ext_vector_type(2)));
typedef float v4f __attribute__((ext_vector_type(4)));
typedef float v8f __attribute__((ext_vector_type(8)));

#define NQ  4
#define DIM 16
#define PPI 196   // patches per image (14*14)

// DS_SWIZZLE_B32 group-of-32 XOR exchange: offset = and | (or<<5) | (xor<<10).
template <int XMASK>
__device__ __forceinline__ float swz_xor(float v) {
  return __int_as_float(
      __builtin_amdgcn_ds_swizzle(__float_as_int(v), (XMASK << 10) | 0x1F));
}

// ---------------------------------------------------------------------------
// Kernel 1: build the fixed 16x16 real operator U(theta) (CNOT-ring + RY
// layers) by evolving each of the 16 basis states. U[n][t] = <n|U|t>.
// qubit q <-> bit (3-q) of the index (axis 1 of [2,2,2,2] is MSB).
// ---------------------------------------------------------------------------
__global__ void build_unitary_kernel(const float* __restrict__ theta, int depth,
                                     float* __restrict__ U) {
  int t = threadIdx.x;
  if (t >= DIM) return;
  float st[DIM];
#pragma unroll
  for (int n = 0; n < DIM; ++n) st[n] = (n == t) ? 1.f : 0.f;
  for (int d = 0; d < depth; ++d) {
#pragma unroll
    for (int q = 0; q < NQ; ++q) {           // CNOT ring q -> (q+1)%4
      int cm = 8 >> q;
      int gm = 8 >> ((q + 1) & 3);
      float tmp[DIM];
#pragma unroll
      for (int n = 0; n < DIM; ++n) tmp[n] = (n & cm) ? st[n ^ gm] : st[n];
#pragma unroll
      for (int n = 0; n < DIM; ++n) st[n] = tmp[n];
    }
#pragma unroll
    for (int q = 0; q < NQ; ++q) {           // RY(theta[d][q])
      float h = 0.5f * theta[d * NQ + q];
      float c = __cosf(h), s = __sinf(h);
      int qm = 8 >> q;
#pragma unroll
      for (int n = 0; n < DIM; ++n) {
        if (!(n & qm)) {
          float s0 = st[n], s1 = st[n | qm];
          st[n]      = c * s0 - s * s1;
          st[n | qm] = s * s0 + c * s1;
        }
      }
    }
  }
  for (int n = 0; n < DIM; ++n) U[n * DIM + t] = st[n];
}

// ---------------------------------------------------------------------------
// Kernel 2: conv(2x2,s2) -> product-state v -> psi = U v (4 chained WMMAs)
// -> square -> feats = psi^2 * SignMatrix via a SECOND WMMA chain whose B
// operand is pure register constants (+/-1, 0). The C/D->A layout mismatch
// is bridged by a per-wave 16x16 LDS transpose (8 stores + 4 2addr loads).
// Block = 256 threads = 8 waves; each wave owns 32 patches (2 WMMA tiles).
// ---------------------------------------------------------------------------
__global__ void __launch_bounds__(256)
quanv_kernel(const float* __restrict__ x,    // [B,1,28,28]
             const float* __restrict__ cw,   // [4,1,2,2]
             const float* __restrict__ cb,   // [4]
             const float* __restrict__ U,    // [16][16]
             float* __restrict__ feats) {    // [B*196][4] == [B][784]
  __shared__ float v_lds[8][32][DIM + 1];    // encoded product states
  __shared__ float p_lds[8][DIM][DIM + 1];   // per-wave psi^2 transpose tile
  __shared__ float f_lds[8][32][NQ];         // staged feats (block-contiguous)
  __shared__ float dump[8];

  const int wave = threadIdx.x >> 5;
  const int lane = threadIdx.x & 31;
  const int wave_patch0 = (blockIdx.x * 8 + wave) * 32;

  // ---- phase 1: one patch per lane (all 32 lanes active) ----
  {
    int p   = wave_patch0 + lane;
    int b   = p / PPI;
    int rem = p - b * PPI;
    int r   = rem / 14, c = rem - r * 14;
    const float* xb = x + ((size_t)b * 28 + 2 * r) * 28 + 2 * c;
    float p00 = xb[0], p01 = xb[1], p10 = xb[28], p11 = xb[29];
    float cs[NQ], sn[NQ];
#pragma unroll
    for (int q = 0; q < NQ; ++q) {
      const float* w = cw + 4 * q;
      float ang = cb[q] + w[0] * p00 + w[1] * p01 + w[2] * p10 + w[3] * p11;
      __sincosf(0.5f * ang, &sn[q], &cs[q]);
    }
    float ab[4], cd[4];                      // outer-product tree: 24 mults
    ab[0] = cs[0] * cs[1]; ab[1] = cs[0] * sn[1];
    ab[2] = sn[0] * cs[1]; ab[3] = sn[0] * sn[1];
    cd[0] = cs[2] * cs[3]; cd[1] = cs[2] * sn[3];
    cd[2] = sn[2] * cs[3]; cd[3] = sn[2] * sn[3];
#pragma unroll
    for (int n = 0; n < DIM; ++n)
      v_lds[wave][lane][n] = ab[n >> 2] * cd[n & 3];
  }
  __syncthreads();

  const int m  = lane & 15;
  const int hi = lane >> 4;

  // B fragment of U^T (16x16x4 layout), loaded once; reused for both tiles.
  v2f ub[4];
#pragma unroll
  for (int k = 0; k < 4; ++k)
    ub[k] = *(const v2f*)(U + m * DIM + 4 * k + 2 * hi);

  // Sign-matrix B fragments, pure constants: S[n][q] = (-1)^{bit_(3-q)(n)}
  // for q<4, else 0. Lane's column is q = lane&15; rows are n = 4c+2hi,+1.
  const bool qvalid = (m < 4);
  const int  sh     = qvalid ? (3 - m) : 0;
  v2f sb[4];
#pragma unroll
  for (int c = 0; c < 4; ++c) {
    int n0 = 4 * c + 2 * hi;
    sb[c].x = qvalid ? (((n0 >> sh) & 1) ? -1.f : 1.f) : 0.f;
    sb[c].y = qvalid ? ((((n0 + 1) >> sh) & 1) ? -1.f : 1.f) : 0.f;
  }

#pragma unroll
  for (int t = 0; t < 2; ++t) {
    // GEMM 1: psi[m][n] = sum_k v[m][k] * U[n][k]
    v8f acc = {};
#pragma unroll
    for (int k = 0; k < 4; ++k) {
      int koff = 4 * k + 2 * hi;
      v2f a;
      a.x = v_lds[wave][t * 16 + m][koff];
      a.y = v_lds[wave][t * 16 + m][koff + 1];
      acc = __builtin_amdgcn_wmma_f32_16x16x4_f32(
          false, a, false, ub[k], (short)0, acc, false, false);
    }

    // transpose psi^2 from C/D layout into A layout via per-wave LDS tile
    __builtin_amdgcn_wave_barrier();
#pragma unroll
    for (int j = 0; j < 8; ++j)
      p_lds[wave][j + 8 * hi][m] = acc[j] * acc[j];
    __builtin_amdgcn_wave_barrier();

    // GEMM 2: feats[m][q] = sum_n psi2[m][n] * S[n][q]  (B = reg constants)
    v8f fz = {};
#pragma unroll
    for (int c = 0; c < 4; ++c) {
      int koff = 4 * c + 2 * hi;
      v2f a;
      a.x = p_lds[wave][m][koff];
      a.y = p_lds[wave][m][koff + 1];
      fz = __builtin_amdgcn_wmma_f32_16x16x4_f32(
          false, a, false, sb[c], (short)0, fz, false, false);
    }
    __builtin_amdgcn_wave_barrier();   // p_lds reused next tile

    // D layout: VGPR j -> patch m=j+8*hi, col q=lane&15 (valid q<4):
    // exactly the [patch][q] order -- branchless staging store.
#pragma unroll
    for (int j = 0; j < 8; ++j) {
      float* dst = qvalid ? &f_lds[wave][t * 16 + j + 8 * hi][m] : &dump[wave];
      *dst = fz[j];
    }
  }
  __syncthreads();

  // coalesced store: block covers 256 patches * 4 feats = 1024 contiguous f32
  const v4f* src = (const v4f*)&f_lds[0][0][0];
  v4f* dstg = (v4f*)(feats + (size_t)blockIdx.x * 1024);
  dstg[threadIdx.x] = src[threadIdx.x];
}

// ---------------------------------------------------------------------------
// Kernel 3: logits = feats[B,784] @ fc_w^T + fc_b, then log_softmax.
// WMMA GEMM: M=16 images/wave, N=16 (10 valid, zero-padded in LDS), K=784.
// Softmax reductions via ds_swizzle; logits staged in LDS -> coalesced f4 out.
// ---------------------------------------------------------------------------
__global__ void __launch_bounds__(256)
fc_logsoftmax_kernel(const float* __restrict__ feats,  // [B][784]
                     const float* __restrict__ fcw,    // [10][784]
                     const float* __restrict__ fcb,    // [10]
                     float* __restrict__ out) {        // [B][10]
  __shared__ float wt[DIM][785];            // 785: conflict-free row stride
  __shared__ float lg[128][10];             // staged logits
  __shared__ float dump;
  for (int i = threadIdx.x; i < DIM * 784; i += 256) {
    int n = i / 784, k = i - n * 784;
    wt[n][k] = (n < 10) ? fcw[n * 784 + k] : 0.f;
  }
  __syncthreads();

  const int wave = threadIdx.x >> 5;
  const int lane = threadIdx.x & 31;
  const int m  = lane & 15;                 // image row (A) / class col (B)
  const int hi = lane >> 4;
  const int img_base = blockIdx.x * 128 + wave * 16;
  const float* arow = feats + (size_t)(img_base + m) * 784 + 2 * hi;
  const float* wrow = &wt[m][2 * hi];

  v8f acc = {};
#pragma unroll 4
  for (int k = 0; k < 784; k += 4) {
    v2f a = *(const v2f*)(arow + k);        // 8B-aligned (784 even, k even)
    v2f b;
    b.x = wrow[k];
    b.y = wrow[k + 1];
    acc = __builtin_amdgcn_wmma_f32_16x16x4_f32(
        false, a, false, b, (short)0, acc, false, false);
  }

  const int  n     = m;
  const bool valid = (n < 10);
  float bias = valid ? fcb[n] : 0.f;
#pragma unroll
  for (int j = 0; j < 8; ++j) {
    float logit = acc[j] + bias;
    float xm = valid ? logit : -INFINITY;
    xm = fmaxf(xm, swz_xor<1>(xm));
    xm = fmaxf(xm, swz_xor<2>(xm));
    xm = fmaxf(xm, swz_xor<4>(xm));
    xm = fmaxf(xm, swz_xor<8>(xm));
    float e = valid ? __expf(logit - xm) : 0.f;
    e += swz_xor<1>(e);
    e += swz_xor<2>(e);
    e += swz_xor<4>(e);
    e += swz_xor<8>(e);
    float lse = xm + __logf(e);
    float* dst = valid ? &lg[wave * 16 + j + 8 * hi][n] : &dump;
    *dst = logit - lse;
  }
  __syncthreads();

  // 128 images * 10 logits = 1280 floats = 320 float4, coalesced
  const v4f* src = (const v4f*)&lg[0][0];
  v4f* dstg = (v4f*)(out + (size_t)blockIdx.x * 1280);
  for (int i = threadIdx.x; i < 320; i += 256)
    dstg[i] = src[i];
}

// ---------------------------------------------------------------------------
extern "C" void kernel_launch(void* const* d_in, const int* in_sizes, int n_in,
                              void* d_out, int out_size, void* d_ws, size_t ws_size,
                              hipStream_t stream) {
  const float* x     = (const float*)d_in[0];   // [B,1,28,28]
  const float* convw = (const float*)d_in[1];   // [4,1,2,2]
  const float* convb = (const float*)d_in[2];   // [4]
  const float* theta = (const float*)d_in[3];   // [depth,4]
  const float* fcw   = (const float*)d_in[4];   // [10,784]
  const float* fcb   = (const float*)d_in[5];   // [10]
  float* out = (float*)d_out;

  float* U     = (float*)d_ws;                  // 256 floats
  float* feats = (float*)((char*)d_ws + 1024);  // B*784 floats, 16B aligned

  const int B     = in_sizes[0] / (28 * 28);    // 4096
  const int depth = in_sizes[3] / NQ;           // 2
  const int P     = B * PPI;                    // 802816 = 3136 * 256

  build_unitary_kernel<<<1, 32, 0, stream>>>(theta, depth, U);
  quanv_kernel<<<P / 256, 256, 0, stream>>>(x, convw, convb, U, feats);
  fc_logsoftmax_kernel<<<B / 128, 256, 0, stream>>>(feats, fcw, fcb, out);
}